// ReactorNet_62758062129174
// MI455X (gfx1250) — compile-verified
//
#include <hip/hip_runtime.h>

// ---------------- problem constants (from reference setup) ----------------
#define S        53
#define SP       56
#define RXN      325
#define NSTATE   54
#define NP       64            // padded matrix dim (identity padding)
#define LDA      65            // LDS row stride (bank-conflict free columns)
#define DT_F     1e-8f
#define NSTEPS   25
#define NNEWTON  3
#define RGAS     8.314462618f

typedef float v8f __attribute__((ext_vector_type(8)));
typedef float v2f __attribute__((ext_vector_type(2)));

// -------- prep: extract 2+2 reaction slots (rows of nu_f/nu_b sum to 2) ---
__global__ void prep_slots_kernel(const int* __restrict__ nu_f,
                                  const int* __restrict__ nu_b,
                                  int4* __restrict__ slots) {
  int r = blockIdx.x * blockDim.x + threadIdx.x;
  if (r >= RXN) return;
  int f0 = 0, f1 = 0, p0 = 0, p1 = 0, nf = 0, npr = 0;
  for (int s = 0; s < S; ++s) {
    int cf = nu_f[r * S + s];
    while (cf-- > 0) { if (nf == 0) f0 = s; else f1 = s; ++nf; }
    int cb = nu_b[r * S + s];
    while (cb-- > 0) { if (npr == 0) p0 = s; else p1 = s; ++npr; }
  }
  slots[r] = make_int4(f0, f1, p0, p1);
}

// ---------------- main: one workgroup (2 wave32) per reactor --------------
__global__ __launch_bounds__(64, 1) void reactor_kernel(
    const float* __restrict__ T0g, const float* __restrict__ Y0g,
    const float* __restrict__ Pg,  const float* __restrict__ Wg,
    const float* __restrict__ nLg, const float* __restrict__ nHg,
    const float* __restrict__ Tmg, const float* __restrict__ Ag,
    const float* __restrict__ bg,  const float* __restrict__ Eag,
    const int4*  __restrict__ slotsg, float* __restrict__ out)
{
  __shared__ float Msh[NP * LDA];                 // M = I - DT*J, then LU factors
  __shared__ float ycur[NP], ynew[NP], fv[NP], rv[NP], xv[NP];
  __shared__ float Wl[SP], invWl[SP], yowl[SP], Cl[SP], maskl[SP];
  __shared__ float cpRl[SP], dcpRl[SP], hRTl[SP], dhRTl[SP];
  __shared__ float gRTl[SP], dgRTl[SP], hmoll[SP], wdl[SP], Tml[SP];
  __shared__ float nLl[S * 7], nHl[S * 7];
  __shared__ float Al[RXN], bl[RXN], Eal[RXN], kfl[RXN], krl[RXN];
  __shared__ int4  sl[RXN];
  __shared__ float scal[8];                       // 0:Wbar 1:rho 2:cp_mass 3:sumhw 4:P

  const int tid  = threadIdx.x;
  const int b    = blockIdx.x;
  const int lane = tid & 31;
  const int wv   = tid >> 5;

  // ---- stage constants into LDS (reused 75x) ----
  for (int i = tid; i < S * 7; i += 64) { nLl[i] = nLg[i]; nHl[i] = nHg[i]; }
  for (int i = tid; i < S; i += 64) {
    float w = Wg[i]; Wl[i] = w; invWl[i] = 1.f / w; Tml[i] = Tmg[i];
  }
  for (int r = tid; r < RXN; r += 64) {
    Al[r] = Ag[r]; bl[r] = bg[r]; Eal[r] = Eag[r]; sl[r] = slotsg[r];
  }
  if (tid == 0) scal[4] = Pg[0];
  {
    float v = 0.f;
    if (tid == 0) v = T0g[b];
    else if (tid <= S) v = Y0g[b * S + (tid - 1)];
    ycur[tid] = v; ynew[tid] = v;
  }
  __syncthreads();
  const float P = scal[4];

  for (int step = 0; step < NSTEPS; ++step) {
    for (int it = 0; it < NNEWTON; ++it) {
      // ================= primal RHS at ynew =================
      const float T    = ynew[0];
      const float invT = 1.f / T;
      const float lnT  = __logf(T);

      if (tid < S) yowl[tid] = ynew[1 + tid] * invWl[tid];
      __syncthreads();
      if (tid == 0) {
        float s = 0.f;
        for (int i = 0; i < S; ++i) s += yowl[i];
        float Wbar = 1.f / s;
        scal[0] = Wbar;
        scal[1] = P * Wbar / (RGAS * T);
      }
      __syncthreads();
      const float rho  = scal[1];
      const float Wbar = scal[0];

      if (tid < S) {
        const int i = tid;
        float u = rho * yowl[i];
        maskl[i] = (u > 1e-12f) ? 1.f : 0.f;
        Cl[i] = fmaxf(u, 1e-12f);
        const float* a = (T < Tml[i]) ? &nLl[i * 7] : &nHl[i * 7];
        float a0=a[0],a1=a[1],a2=a[2],a3=a[3],a4=a[4],a5=a[5],a6=a[6];
        float T2 = T*T, T3 = T2*T, T4 = T3*T;
        float cp  = a0 + a1*T + a2*T2 + a3*T3 + a4*T4;
        float h   = a0 + a1*T*0.5f + a2*T2*(1.f/3.f) + a3*T3*0.25f + a4*T4*0.2f + a5*invT;
        float sR  = a0*lnT + a1*T + a2*T2*0.5f + a3*T3*(1.f/3.f) + a4*T4*0.25f + a6;
        float dcp = a1 + 2.f*a2*T + 3.f*a3*T2 + 4.f*a4*T3;
        float dh  = a1*0.5f + a2*T*(2.f/3.f) + a3*T2*0.75f + a4*T3*0.8f - a5*invT*invT;
        float ds  = a0*invT + a1 + a2*T + a3*T2 + a4*T3;
        cpRl[i]=cp; dcpRl[i]=dcp; hRTl[i]=h; dhRTl[i]=dh;
        gRTl[i]=h - sR; dgRTl[i]=dh - ds; hmoll[i]=h * RGAS * T;
        wdl[i] = 0.f;
      }
      __syncthreads();
      if (tid == 1) {                       // cp_mass
        float s = 0.f;
        for (int i = 0; i < S; ++i) s += ynew[1 + i] * cpRl[i] * invWl[i];
        scal[2] = s * RGAS;
      }
      for (int i = tid; i < NP * LDA; i += 64) Msh[i] = 0.f;   // column scratch
      __syncthreads();

      // reactions: A+B -> C+D with dnu==0  =>  Kc = exp(-dG), kr = kf*exp(dG)
      for (int r = tid; r < RXN; r += 64) {
        int4 s4 = sl[r];
        float kf = Al[r] * __expf(bl[r] * lnT - Eal[r] * invT);
        float dG = gRTl[s4.z] + gRTl[s4.w] - gRTl[s4.x] - gRTl[s4.y];
        float kr = kf * __expf(dG);
        float q  = kf * Cl[s4.x] * Cl[s4.y] - kr * Cl[s4.z] * Cl[s4.w];
        kfl[r] = kf; krl[r] = kr;
        atomicAdd(&wdl[s4.x], -q);          // ds_add_f32
        atomicAdd(&wdl[s4.y], -q);
        atomicAdd(&wdl[s4.z],  q);
        atomicAdd(&wdl[s4.w],  q);
      }
      __syncthreads();
      const float cp_mass = scal[2];
      if (tid < S) fv[1 + tid] = wdl[tid] * Wl[tid] / rho;
      if (tid == 0) {
        float s = 0.f;
        for (int i = 0; i < S; ++i) s += hmoll[i] * wdl[i];
        scal[3] = s;
        fv[0] = -s / (rho * cp_mass);
      }
      if (tid >= NSTATE) fv[tid] = 0.f;
      __syncthreads();
      const float sumhw = scal[3];

      // ============ analytic forward-mode Jacobian: one column per thread ===
      if (tid < NSTATE) {
        const int j = tid, jj = j - 1;
        const float drho = (j == 0) ? (-rho * invT) : (-rho * Wbar * invWl[jj]);
        for (int r = 0; r < RXN; ++r) {
          int4 s4 = sl[r];
          float kf = kfl[r], kr = krl[r];
          float Ca = Cl[s4.x], Cb = Cl[s4.y], Cc = Cl[s4.z], Cd = Cl[s4.w];
          float dCa = maskl[s4.x] * (drho * yowl[s4.x] + ((j > 0 && s4.x == jj) ? rho * invWl[s4.x] : 0.f));
          float dCb = maskl[s4.y] * (drho * yowl[s4.y] + ((j > 0 && s4.y == jj) ? rho * invWl[s4.y] : 0.f));
          float dCc = maskl[s4.z] * (drho * yowl[s4.z] + ((j > 0 && s4.z == jj) ? rho * invWl[s4.z] : 0.f));
          float dCd = maskl[s4.w] * (drho * yowl[s4.w] + ((j > 0 && s4.w == jj) ? rho * invWl[s4.w] : 0.f));
          float dCf = dCa * Cb + Ca * dCb;
          float dCr = dCc * Cd + Cc * dCd;
          float dq;
          if (j == 0) {
            float dlkf = bl[r] * invT + Eal[r] * invT * invT;
            float ddG  = dgRTl[s4.z] + dgRTl[s4.w] - dgRTl[s4.x] - dgRTl[s4.y];
            dq = kf * dlkf * (Ca * Cb) + kf * dCf
               - kr * (dlkf + ddG) * (Cc * Cd) - kr * dCr;
          } else {
            dq = kf * dCf - kr * dCr;
          }
          Msh[(1 + s4.x) * LDA + j] -= dq;
          Msh[(1 + s4.y) * LDA + j] -= dq;
          Msh[(1 + s4.z) * LDA + j] += dq;
          Msh[(1 + s4.w) * LDA + j] += dq;
        }
        // column post-transform: dwdot -> rows of (I - DT*J)
        float dcp, sumdh_w = 0.f;
        if (j == 0) {
          float sc = 0.f, sw = 0.f;
          for (int i = 0; i < S; ++i) {
            sc += ynew[1 + i] * dcpRl[i] * invWl[i];
            sw += (hRTl[i] + T * dhRTl[i]) * wdl[i];
          }
          dcp = sc * RGAS; sumdh_w = sw * RGAS;
        } else {
          dcp = cpRl[jj] * RGAS * invWl[jj];
        }
        float sumh_dw = 0.f;
        const float invrho = 1.f / rho;
        for (int i = 0; i < S; ++i) {
          float dw = Msh[(1 + i) * LDA + j];
          sumh_dw += hmoll[i] * dw;
          float val = Wl[i] * invrho * dw - fv[1 + i] * drho * invrho;
          Msh[(1 + i) * LDA + j] = (((1 + i) == j) ? 1.f : 0.f) - DT_F * val;
        }
        float den  = rho * cp_mass;
        float dden = drho * cp_mass + rho * dcp;
        float dTdt_d = -((sumh_dw + sumdh_w) * den - sumhw * dden) / (den * den);
        Msh[j] = ((j == 0) ? 1.f : 0.f) - DT_F * dTdt_d;
      } else {
        Msh[tid * LDA + tid] = 1.f;                 // identity padding
      }
      rv[tid] = (tid < NSTATE) ? (ynew[tid] - ycur[tid] - DT_F * fv[tid]) : 0.f;
      __syncthreads();

      // ======== blocked LU (no pivot; M ~ I) with WMMA trailing updates =====
      for (int kb = 0; kb < NP; kb += 16) {
        for (int jp = kb; jp < kb + 16; ++jp) {     // panel factorization
          __syncthreads();
          float invp = 1.f / Msh[jp * LDA + jp];
          if (tid > jp) {
            float m = Msh[tid * LDA + jp] * invp;
            Msh[tid * LDA + jp] = m;
            for (int c = jp + 1; c < kb + 16; ++c)
              Msh[tid * LDA + c] -= m * Msh[jp * LDA + c];
          }
        }
        __syncthreads();
        const int ncols = NP - kb - 16;
        if (ncols > 0) {
          if (tid < ncols) {                        // U12 = L11^{-1} A12 (per column)
            int c = kb + 16 + tid;
            for (int i2 = kb; i2 < kb + 15; ++i2) {
              float v = Msh[i2 * LDA + c];
              for (int j2 = i2 + 1; j2 < kb + 16; ++j2)
                Msh[j2 * LDA + c] -= Msh[j2 * LDA + i2] * v;
            }
          }
          __syncthreads();
          const int nb = ncols >> 4;                // A22 -= L21*U12 via WMMA f32
          for (int t = wv; t < nb * nb; t += 2) {
            const int I  = kb + 16 + (t / nb) * 16;
            const int J  = kb + 16 + (t % nb) * 16;
            const int m  = lane & 15;
            const int hp = lane >> 4;               // half-wave select
            v8f c;
            #pragma unroll
            for (int g = 0; g < 8; ++g)
              c[g] = Msh[(I + g + hp * 8) * LDA + (J + m)];
            #pragma unroll
            for (int kk = 0; kk < 4; ++kk) {
              const int kc = kb + 4 * kk;
              v2f av, bv;
              av[0] = -Msh[(I + m) * LDA + (kc + 2 * hp)];      // negate A -> subtract
              av[1] = -Msh[(I + m) * LDA + (kc + 2 * hp + 1)];
              bv[0] =  Msh[(kc + 2 * hp) * LDA + (J + m)];
              bv[1] =  Msh[(kc + 2 * hp + 1) * LDA + (J + m)];
#if defined(__gfx1250__) && __has_builtin(__builtin_amdgcn_wmma_f32_16x16x4_f32)
              c = __builtin_amdgcn_wmma_f32_16x16x4_f32(
                      false, av, false, bv, (short)0, c, false, false);
#else
              #pragma unroll
              for (int g = 0; g < 8; ++g) {         // scalar fallback
                float acc = c[g];
                for (int k2 = 0; k2 < 4; ++k2)
                  acc -= Msh[(I + g + hp * 8) * LDA + (kc + k2)] *
                         Msh[(kc + k2) * LDA + (J + m)];
                c[g] = acc;
              }
#endif
            }
            #pragma unroll
            for (int g = 0; g < 8; ++g)
              Msh[(I + g + hp * 8) * LDA + (J + m)] = c[g];
          }
          __syncthreads();
        }
      }
      // forward substitution (unit lower)
      for (int jp = 0; jp < NP; ++jp) {
        __syncthreads();
        float xj = rv[jp];
        if (tid > jp) rv[tid] -= Msh[tid * LDA + jp] * xj;
      }
      // backward substitution
      for (int jp = NP - 1; jp >= 0; --jp) {
        __syncthreads();
        float xj = rv[jp] / Msh[jp * LDA + jp];
        if (tid == jp) xv[jp] = xj;
        if (tid < jp) rv[tid] -= Msh[tid * LDA + jp] * xj;
      }
      __syncthreads();
      if (tid < NSTATE) ynew[tid] -= xv[tid];       // Newton update
      __syncthreads();
    }
    if (tid < NSTATE) ycur[tid] = ynew[tid];        // commit BE step
    __syncthreads();
  }
  if (tid < NSTATE) out[b * NSTATE + tid] = ycur[tid];
}

extern "C" void kernel_launch(void* const* d_in, const int* in_sizes, int n_in,
                              void* d_out, int out_size, void* d_ws, size_t ws_size,
                              hipStream_t stream) {
  (void)n_in; (void)out_size; (void)ws_size;
  const float* T0 = (const float*)d_in[0];
  const float* Y0 = (const float*)d_in[1];
  const float* P  = (const float*)d_in[2];
  const float* W  = (const float*)d_in[3];
  const float* nL = (const float*)d_in[4];
  const float* nH = (const float*)d_in[5];
  const float* Tm = (const float*)d_in[6];
  const float* A  = (const float*)d_in[7];
  const float* bb = (const float*)d_in[8];
  const float* Ea = (const float*)d_in[9];
  const int*   nf = (const int*)d_in[10];
  const int*   nb = (const int*)d_in[11];
  const int    B  = in_sizes[0];

  int4* slots = (int4*)d_ws;
  prep_slots_kernel<<<(RXN + 63) / 64, 64, 0, stream>>>(nf, nb, slots);
  reactor_kernel<<<B, 64, 0, stream>>>(T0, Y0, P, W, nL, nH, Tm, A, bb, Ea,
                                       slots, (float*)d_out);
}